// FashionNetFPN_40097814676079
// MI455X (gfx1250) — compile-verified
//
#include <hip/hip_runtime.h>
#include <hip/hip_bf16.h>
#include <math.h>

#define NIMG 8
#define CCH 256
#define AA 9
#define KK 80
#define TOPK_T 1000
#define TTOT 5000
#define MAXDET 100
#define NCAND 4096
#define NBINS 4096
#define TOTCLS 30
#define FEATD 1280
#define IMGSZ 800.0f
#define SCORE_TH 0.05f
#define NMS_TH 0.5f
#define SCALE_CLAMP_F 4.135166556742356f

typedef __attribute__((ext_vector_type(2))) float v2f;
typedef __attribute__((ext_vector_type(8))) float v8f;
typedef __attribute__((ext_vector_type(4))) unsigned tdm_g0_t;
typedef __attribute__((ext_vector_type(8))) int tdm_g1_t;
typedef __attribute__((ext_vector_type(4))) int tdm_g2_t;
typedef __attribute__((ext_vector_type(8))) int tdm_g4_t;

__device__ __forceinline__ unsigned okey(float f) {
  unsigned u = __float_as_uint(f);
  return (u & 0x80000000u) ? ~u : (u | 0x80000000u);
}

// ---------------- init: zero histograms + scalars/counters ----------------
__global__ void init_kernel(int* __restrict__ hist, int* __restrict__ scal) {
  int i = blockIdx.x * blockDim.x + threadIdx.x;
  if (i < 40 * NBINS) hist[i] = 0;
  if (i < 40 * 8) scal[i] = 0;
}

// ---------------- adaptive avg pool 1x1 per (n, level, channel) ----------------
__global__ __launch_bounds__(256)
void pool_kernel(const float* __restrict__ feat, float* __restrict__ pooled,
                 int HW, int lvl) {
  __shared__ float red[256];
  int n = blockIdx.x / CCH;
  int c = blockIdx.x % CCH;
  const float* p = feat + ((size_t)n * CCH + c) * HW;
  float s = 0.0f;
  for (int i = threadIdx.x; i < HW; i += 256) s += p[i];
  red[threadIdx.x] = s;
  __syncthreads();
  for (int st = 128; st > 0; st >>= 1) {
    if (threadIdx.x < st) red[threadIdx.x] += red[threadIdx.x + st];
    __syncthreads();
  }
  if (threadIdx.x == 0)
    pooled[(size_t)n * FEATD + lvl * CCH + c] = red[0] / (float)HW;
}

// ---------------- classification head: TDM -> LDS -> f32 WMMA ----------------
// A = pooled (8x1280) staged to LDS by the Tensor Data Mover, padded to 16 rows
// via mask-multiplies (no divergent branches: EXEC stays all-ones for WMMA).
// B = linear_w^T (1280x30 -> two 16-col N-tiles). f32 WMMA 16x16x4, K-loop 320.
__global__ __launch_bounds__(32)
void linear_wmma_kernel(const float* __restrict__ pooled,
                        const float* __restrict__ lw,
                        const float* __restrict__ lb,
                        float* __restrict__ out_logits) {
  __shared__ __align__(16) float sA[NIMG * FEATD];  // 40KB
  int lane = threadIdx.x;
  int half = lane >> 4;
  int m = lane & 15;

#if __has_builtin(__builtin_amdgcn_tensor_load_to_lds)
  {
    unsigned lds_base = (unsigned)(size_t)(void*)sA;
    unsigned long long ga = (unsigned long long)(size_t)pooled;
    tdm_g0_t g0;
    g0[0] = 1u;                                    // count=1, user descriptor
    g0[1] = lds_base;                              // lds_addr
    g0[2] = (unsigned)(ga & 0xFFFFFFFFull);        // global_addr lo
    g0[3] = (unsigned)(ga >> 32) | (2u << 30);     // global_addr hi | type=2
    tdm_g1_t g1;
    g1[0] = (int)(2u << 16);                       // data_size=4B, mask=0
    g1[1] = (int)((FEATD & 0xFFFF) << 16);         // tensor_dim0[15:0] @ bit48
    g1[2] = (int)((FEATD >> 16) | ((NIMG & 0xFFFF) << 16));  // dim0 hi | dim1 lo
    g1[3] = (int)((NIMG >> 16) | (FEATD << 16));   // dim1 hi | tile_dim0=1280
    g1[4] = (int)(NIMG & 0xFFFF);                  // tile_dim1=8, tile_dim2=0
    g1[5] = (int)FEATD;                            // tensor_dim0_stride lo
    g1[6] = 0;                                     // stride0 hi | stride1 lo
    g1[7] = 0;
    tdm_g2_t gz2 = {};
    tdm_g2_t gz3 = {};
#if defined(__clang_major__) && __clang_major__ >= 23
    tdm_g4_t gz4 = {};
    __builtin_amdgcn_tensor_load_to_lds(g0, g1, gz2, gz3, gz4, 0);
#else
    __builtin_amdgcn_tensor_load_to_lds(g0, g1, gz2, gz3, 0);
#endif
    __builtin_amdgcn_s_wait_tensorcnt(0);
  }
#else
  for (int i = lane; i < NIMG * FEATD; i += 32) sA[i] = pooled[i];
#endif
  __syncthreads();

  float maskA = (m < NIMG) ? 1.0f : 0.0f;
  int ra = (m < NIMG) ? m : 0;
  int c2 = 16 + m;
  float maskB1 = (c2 < TOTCLS) ? 1.0f : 0.0f;
  int cc2 = (c2 < TOTCLS) ? c2 : 0;

  v8f c0 = {};
  v8f c1 = {};
  for (int k = 0; k < FEATD; k += 4) {
    int kk = k + 2 * half;
    v2f a = *(const v2f*)&sA[ra * FEATD + kk];
    a.x *= maskA; a.y *= maskA;
    v2f b0 = *(const v2f*)&lw[m * FEATD + kk];     // N = m < 16 < 30, always valid
    v2f b1 = *(const v2f*)&lw[cc2 * FEATD + kk];
    b1.x *= maskB1; b1.y *= maskB1;
    c0 = __builtin_amdgcn_wmma_f32_16x16x4_f32(false, a, false, b0, (short)0, c0,
                                               false, false);
    c1 = __builtin_amdgcn_wmma_f32_16x16x4_f32(false, a, false, b1, (short)0, c1,
                                               false, false);
  }
  // D layout: lanes 0-15 hold N=lane, VGPR r -> M=r (rows 0..7 = real images)
  if (half == 0) {
    for (int r = 0; r < NIMG; ++r) {
      out_logits[r * TOTCLS + m] = c0[r] + lb[m];
      if (c2 < TOTCLS) out_logits[r * TOTCLS + c2] = c1[r] + lb[c2];
    }
  }
}

// ---------------- single-pass 12-bit histogram (LDS privatized, b128 loads) ----------------
__global__ __launch_bounds__(256)
void hist_kernel(const float* __restrict__ cls, long M, int lvl,
                 int* __restrict__ hist) {
  __shared__ int lh[NBINS];
  int n = blockIdx.y;
  int id = n * 5 + lvl;
  for (int i = threadIdx.x; i < NBINS; i += 256) lh[i] = 0;
  __syncthreads();
  const float4* p4 = (const float4*)(cls + (size_t)n * M);
  long M4 = M >> 2;
  long stride = (long)gridDim.x * blockDim.x;
  for (long q = (long)blockIdx.x * blockDim.x + threadIdx.x; q < M4; q += stride) {
    float4 v = p4[q];
    atomicAdd(&lh[okey(v.x) >> 20], 1);
    atomicAdd(&lh[okey(v.y) >> 20], 1);
    atomicAdd(&lh[okey(v.z) >> 20], 1);
    atomicAdd(&lh[okey(v.w) >> 20], 1);
  }
  __syncthreads();
  int* g = hist + (size_t)id * NBINS;
  for (int i = threadIdx.x; i < NBINS; i += 256) {
    int v = lh[i];
    if (v) atomicAdd(&g[i], v);
  }
}

// ---------------- pick 12-bit floor bucket per (n,l): count(key>=F) >= 1000 ----------------
__global__ __launch_bounds__(256)
void select_kernel(const int* __restrict__ hist, int* __restrict__ scal) {
  __shared__ int part[256];
  int id = blockIdx.x;
  const int* h = hist + (size_t)id * NBINS;
  int t = threadIdx.x;
  int s = 0;
  int base = t * 16;
  for (int b = 0; b < 16; ++b) s += h[base + b];
  part[t] = s;
  __syncthreads();
  if (t == 0) {
    int acc = 0;
    unsigned F = 0;
    for (int c = 255; c >= 0; --c) {
      if (acc + part[c] >= TOPK_T) {
        for (int b = c * 16 + 15; b >= c * 16; --b) {
          acc += h[b];
          if (acc >= TOPK_T) { F = (unsigned)b << 20; break; }
        }
        break;
      }
      acc += part[c];
    }
    scal[id * 8 + 0] = (int)F;
  }
}

// ---------------- compaction of candidates key >= floor ----------------
__global__ __launch_bounds__(256)
void compact_kernel(const float* __restrict__ cls, long M, int HW, int lvl,
                    int* __restrict__ scal, unsigned long long* __restrict__ cand) {
  int n = blockIdx.y;
  int id = n * 5 + lvl;
  int* sc = scal + id * 8;
  unsigned F = (unsigned)sc[0];
  const float4* p4 = (const float4*)(cls + (size_t)n * M);
  unsigned long long* cd = cand + (size_t)id * NCAND;
  long M4 = M >> 2;
  long stride = (long)gridDim.x * blockDim.x;
  for (long q = (long)blockIdx.x * blockDim.x + threadIdx.x; q < M4; q += stride) {
    float4 v = p4[q];
    float vals[4] = {v.x, v.y, v.z, v.w};
#pragma unroll
    for (int j = 0; j < 4; ++j) {
      unsigned key = okey(vals[j]);
      if (key >= F) {
        int slot = atomicAdd(&sc[5], 1);
        if (slot < NCAND) {
          long i = q * 4 + j;
          int ch = (int)(i / HW);          // a*K + kcls (NCHW layout)
          int pos = (int)(i % HW);         // h*W + w
          int a = ch / KK, kc = ch % KK;
          unsigned pidx = (unsigned)((pos * AA + a) * KK + kc);  // permuted idx
          cd[slot] = ((unsigned long long)key << 32) |
                     (unsigned long long)(0xFFFFFFFFu - pidx);   // tie: low idx wins
        }
      }
    }
  }
}

// ---------------- per-(n,l): bitonic sort candidates, decode top-1000 ----------------
__global__ __launch_bounds__(512)
void topk_kernel(const float* __restrict__ cls, const float* __restrict__ delta,
                 const float* __restrict__ anc, int H, int W, int lvl, long M,
                 const int* __restrict__ scal,
                 const unsigned long long* __restrict__ cand,
                 float* __restrict__ cboxes, float* __restrict__ cscores,
                 int* __restrict__ ccls) {
  __shared__ unsigned long long s[NCAND];
  int n = blockIdx.x;
  int id = n * 5 + lvl;
  int cnt = scal[id * 8 + 5];
  if (cnt > NCAND) cnt = NCAND;
  const unsigned long long* cd = cand + (size_t)id * NCAND;
  int tid = threadIdx.x, nthr = blockDim.x;
  for (int i = tid; i < NCAND; i += nthr) s[i] = (i < cnt) ? cd[i] : 0ULL;
  __syncthreads();
  for (unsigned k = 2; k <= NCAND; k <<= 1) {
    for (unsigned j = k >> 1; j > 0; j >>= 1) {
      for (unsigned i = (unsigned)tid; i < NCAND; i += (unsigned)nthr) {
        unsigned ixj = i ^ j;
        if (ixj > i) {
          unsigned long long x = s[i], y = s[ixj];
          bool upper = ((i & k) == 0);     // descending overall
          if (upper ? (x < y) : (x > y)) { s[i] = y; s[ixj] = x; }
        }
      }
      __syncthreads();
    }
  }
  int HW = H * W;
  for (int r = tid; r < TOPK_T; r += nthr) {
    unsigned long long comp = s[r];
    int outp = n * TTOT + lvl * TOPK_T + r;
    float x1 = 0.f, y1 = 0.f, x2 = 0.f, y2 = 0.f, val = 0.f;
    int kcls = 0;
    if (comp != 0ULL) {
      unsigned pidx = 0xFFFFFFFFu - (unsigned)(comp & 0xFFFFFFFFu);
      int kc = (int)(pidx % KK);
      int aidx = (int)(pidx / KK);
      int a = aidx % AA;
      int pos = aidx / AA;
      float logit = cls[(size_t)n * M + (size_t)(a * KK + kc) * HW + pos];
      val = 1.0f / (1.0f + expf(-logit));
      if (!(val > SCORE_TH)) val = 0.0f;
      kcls = kc;
      size_t dbase = (size_t)n * ((size_t)AA * 4 * HW);
      float dx = delta[dbase + (size_t)(a * 4 + 0) * HW + pos];
      float dy = delta[dbase + (size_t)(a * 4 + 1) * HW + pos];
      float dw = delta[dbase + (size_t)(a * 4 + 2) * HW + pos];
      float dh = delta[dbase + (size_t)(a * 4 + 3) * HW + pos];
      dw = fminf(dw, SCALE_CLAMP_F);
      dh = fminf(dh, SCALE_CLAMP_F);
      float ax1 = anc[(size_t)aidx * 4 + 0], ay1 = anc[(size_t)aidx * 4 + 1];
      float ax2 = anc[(size_t)aidx * 4 + 2], ay2 = anc[(size_t)aidx * 4 + 3];
      float aw = ax2 - ax1, ah = ay2 - ay1;
      float acx = ax1 + 0.5f * aw, acy = ay1 + 0.5f * ah;
      float pcx = dx * aw + acx, pcy = dy * ah + acy;
      float pw = expf(dw) * aw, ph = expf(dh) * ah;
      x1 = fminf(fmaxf(pcx - 0.5f * pw, 0.0f), IMGSZ);
      y1 = fminf(fmaxf(pcy - 0.5f * ph, 0.0f), IMGSZ);
      x2 = fminf(fmaxf(pcx + 0.5f * pw, 0.0f), IMGSZ);
      y2 = fminf(fmaxf(pcy + 0.5f * ph, 0.0f), IMGSZ);
    }
    cboxes[(size_t)outp * 4 + 0] = x1;
    cboxes[(size_t)outp * 4 + 1] = y1;
    cboxes[(size_t)outp * 4 + 2] = x2;
    cboxes[(size_t)outp * 4 + 3] = y2;
    cscores[outp] = val;
    ccls[outp] = kcls;
  }
}

// ---------------- greedy batched NMS, one block per image ----------------
__global__ __launch_bounds__(256)
void nms_kernel(const float* __restrict__ cboxes, const float* __restrict__ cscores,
                const int* __restrict__ ccls, float* __restrict__ dout) {
  __shared__ float live[TTOT];
  __shared__ float rv[256];
  __shared__ int ri[256];
  __shared__ int keep[MAXDET];
  __shared__ float bj[5];
  int n = blockIdx.x;
  int tid = threadIdx.x;
  const float* bx = cboxes + (size_t)n * TTOT * 4;
  const float* sc = cscores + (size_t)n * TTOT;
  const int* cl = ccls + (size_t)n * TTOT;
  for (int m = tid; m < TTOT; m += 256) live[m] = sc[m];
  __syncthreads();
  for (int it = 0; it < MAXDET; ++it) {
    float best = -1.0f;
    int bi = 0x7FFFFFFF;
    for (int m = tid; m < TTOT; m += 256) {
      float v = live[m];
      if (v > best || (v == best && m < bi)) { best = v; bi = m; }
    }
    rv[tid] = best; ri[tid] = bi;
    __syncthreads();
    for (int s2 = 128; s2 > 0; s2 >>= 1) {
      if (tid < s2) {
        float ov = rv[tid + s2]; int oi = ri[tid + s2];
        if (ov > rv[tid] || (ov == rv[tid] && oi < ri[tid])) { rv[tid] = ov; ri[tid] = oi; }
      }
      __syncthreads();
    }
    int j = ri[0];
    if (tid == 0) {
      float s = live[j];
      keep[it] = (s > 0.0f) ? j : -1;
      float off = (float)cl[j] * (IMGSZ + 1.0f);
      float jx1 = bx[j * 4 + 0] + off, jy1 = bx[j * 4 + 1] + off;
      float jx2 = bx[j * 4 + 2] + off, jy2 = bx[j * 4 + 3] + off;
      bj[0] = jx1; bj[1] = jy1; bj[2] = jx2; bj[3] = jy2;
      bj[4] = (jx2 - jx1) * (jy2 - jy1);
    }
    __syncthreads();
    float jx1 = bj[0], jy1 = bj[1], jx2 = bj[2], jy2 = bj[3], ja = bj[4];
    for (int m = tid; m < TTOT; m += 256) {
      float off = (float)cl[m] * (IMGSZ + 1.0f);
      float mx1 = bx[m * 4 + 0] + off, my1 = bx[m * 4 + 1] + off;
      float mx2 = bx[m * 4 + 2] + off, my2 = bx[m * 4 + 3] + off;
      float ix1 = fmaxf(jx1, mx1), iy1 = fmaxf(jy1, my1);
      float ix2 = fminf(jx2, mx2), iy2 = fminf(jy2, my2);
      float inter = fmaxf(ix2 - ix1, 0.0f) * fmaxf(iy2 - iy1, 0.0f);
      float am = (mx2 - mx1) * (my2 - my1);
      float iou = inter / (ja + am - inter + 1e-9f);
      if (iou > NMS_TH || m == j) live[m] = 0.0f;
    }
    __syncthreads();
  }
  for (int i = tid; i < MAXDET; i += 256) {
    int k = keep[i];
    bool valid = k >= 0;
    int kc = valid ? k : 0;
    float vm = valid ? 1.0f : 0.0f;
    dout[((size_t)(n * MAXDET + i)) * 4 + 0] = bx[kc * 4 + 0] * vm;
    dout[((size_t)(n * MAXDET + i)) * 4 + 1] = bx[kc * 4 + 1] * vm;
    dout[((size_t)(n * MAXDET + i)) * 4 + 2] = bx[kc * 4 + 2] * vm;
    dout[((size_t)(n * MAXDET + i)) * 4 + 3] = bx[kc * 4 + 3] * vm;
    dout[3200 + n * MAXDET + i] = sc[kc] * vm;
    dout[4000 + n * MAXDET + i] = valid ? (float)cl[kc] : -1.0f;
  }
}

extern "C" void kernel_launch(void* const* d_in, const int* in_sizes, int n_in,
                              void* d_out, int out_size, void* d_ws, size_t ws_size,
                              hipStream_t stream) {
  static const int LH[5] = {100, 50, 25, 13, 7};
  static const int LWd[5] = {100, 50, 25, 13, 7};
  // setup_inputs dict order is interleaved per level: feat, cls, delta, anc
  const float* feat[5];
  const float* cls[5];
  const float* del[5];
  const float* anc[5];
  for (int l = 0; l < 5; ++l) {
    feat[l] = (const float*)d_in[4 * l + 0];
    cls[l]  = (const float*)d_in[4 * l + 1];
    del[l]  = (const float*)d_in[4 * l + 2];
    anc[l]  = (const float*)d_in[4 * l + 3];
  }
  const float* lw = (const float*)d_in[20];
  const float* lb = (const float*)d_in[21];
  float* out = (float*)d_out;

  size_t off = 0;
  char* ws = (char*)d_ws;
  float* pooled = (float*)(ws + off); off += (size_t)NIMG * FEATD * 4;
  int* hist = (int*)(ws + off);       off += (size_t)40 * NBINS * 4;
  int* scal = (int*)(ws + off);       off += (size_t)40 * 8 * 4;
  off = (off + 7) & ~(size_t)7;
  unsigned long long* cand = (unsigned long long*)(ws + off);
  off += (size_t)40 * NCAND * 8;
  float* cboxes = (float*)(ws + off);  off += (size_t)NIMG * TTOT * 4 * 4;
  float* cscores = (float*)(ws + off); off += (size_t)NIMG * TTOT * 4;
  int* ccls = (int*)(ws + off);        off += (size_t)NIMG * TTOT * 4;

  init_kernel<<<(40 * NBINS + 255) / 256, 256, 0, stream>>>(hist, scal);

  for (int l = 0; l < 5; ++l)
    pool_kernel<<<NIMG * CCH, 256, 0, stream>>>(feat[l], pooled, LH[l] * LWd[l], l);

  linear_wmma_kernel<<<1, 32, 0, stream>>>(pooled, lw, lb, out + 4800);

  for (int l = 0; l < 5; ++l) {
    long M = (long)LH[l] * LWd[l] * AA * KK;
    long M4 = M >> 2;
    int gx = (int)((M4 + 32767) / 32768);
    if (gx < 1) gx = 1;
    if (gx > 64) gx = 64;
    hist_kernel<<<dim3(gx, NIMG), 256, 0, stream>>>(cls[l], M, l, hist);
  }

  select_kernel<<<40, 256, 0, stream>>>(hist, scal);

  for (int l = 0; l < 5; ++l) {
    long M = (long)LH[l] * LWd[l] * AA * KK;
    long M4 = M >> 2;
    int gx = (int)((M4 + 32767) / 32768);
    if (gx < 1) gx = 1;
    if (gx > 64) gx = 64;
    compact_kernel<<<dim3(gx, NIMG), 256, 0, stream>>>(cls[l], M, LH[l] * LWd[l], l,
                                                       scal, cand);
  }

  for (int l = 0; l < 5; ++l) {
    long M = (long)LH[l] * LWd[l] * AA * KK;
    topk_kernel<<<NIMG, 512, 0, stream>>>(cls[l], del[l], anc[l], LH[l], LWd[l], l, M,
                                          scal, cand, cboxes, cscores, ccls);
  }

  nms_kernel<<<NIMG, 256, 0, stream>>>(cboxes, cscores, ccls, out);
}